// GATModel_71717363908806
// MI455X (gfx1250) — compile-verified
//
#include <hip/hip_runtime.h>
#include <hip/hip_bf16.h>

#define F_IN 128
#define HID  64
#define NEG_SLOPE 0.2f

typedef float v2f __attribute__((ext_vector_type(2)));
typedef float v8f __attribute__((ext_vector_type(8)));

// ---------------------------------------------------------------------------
// Kernel 1: h = x @ W  via V_WMMA_F32_16X16X4_F32.
// One wave -> 16 rows x 64 cols (4 accumulators). W staged in LDS (32KB).
// ---------------------------------------------------------------------------
__global__ __launch_bounds__(256) void gat_gemm_wmma(
    const float* __restrict__ x, const float* __restrict__ W,
    float* __restrict__ h, int nrows)
{
    __shared__ float Wl[F_IN][HID];   // [k][n], 32 KB
    const int t = threadIdx.x;
    for (int i = t; i < F_IN * HID; i += 256)
        Wl[i >> 6][i & 63] = W[i];
    __syncthreads();

    const int wave = t >> 5;
    const int lane = t & 31;
    const int half = lane >> 4;   // 0: K=+0/+1  1: K=+2/+3
    const int lid  = lane & 15;

    const int rowBase = blockIdx.x * 128 + wave * 16;
    int arow = rowBase + lid;
    if (arow >= nrows) arow = nrows - 1;          // clamp (stores are guarded)
    const float* __restrict__ ap = x + (size_t)arow * F_IN;
    __builtin_prefetch(ap, 0, 1);                  // global_prefetch_b8

    v8f acc0 = {}, acc1 = {}, acc2 = {}, acc3 = {};

    #pragma unroll 4
    for (int kk = 0; kk < F_IN; kk += 4) {
        const int ka = kk + 2 * half;
        v2f a;
        a.x = ap[ka];
        a.y = ap[ka + 1];

        v2f b0, b1, b2, b3;
        b0.x = Wl[ka][lid];          b0.y = Wl[ka + 1][lid];
        b1.x = Wl[ka][16 + lid];     b1.y = Wl[ka + 1][16 + lid];
        b2.x = Wl[ka][32 + lid];     b2.y = Wl[ka + 1][32 + lid];
        b3.x = Wl[ka][48 + lid];     b3.y = Wl[ka + 1][48 + lid];

        acc0 = __builtin_amdgcn_wmma_f32_16x16x4_f32(false, a, false, b0, (short)0, acc0, false, false);
        acc1 = __builtin_amdgcn_wmma_f32_16x16x4_f32(false, a, false, b1, (short)0, acc1, false, false);
        acc2 = __builtin_amdgcn_wmma_f32_16x16x4_f32(false, a, false, b2, (short)0, acc2, false, false);
        acc3 = __builtin_amdgcn_wmma_f32_16x16x4_f32(false, a, false, b3, (short)0, acc3, false, false);
    }

    // C/D layout: VGPR v holds row (v + 8*half), col = lid (+16*tile)
    #pragma unroll
    for (int v = 0; v < 8; ++v) {
        const int gr = rowBase + 8 * half + v;
        if (gr < nrows) {
            float* __restrict__ hp = h + (size_t)gr * HID;
            hp[lid]      = acc0[v];
            hp[16 + lid] = acc1[v];
            hp[32 + lid] = acc2[v];
            hp[48 + lid] = acc3[v];
        }
    }
}

// ---------------------------------------------------------------------------
// Kernel 2: per-node attention scalars + zero-init of softmax scratch.
// One wave per node (wave32; lane handles channels lane, lane+32).
// ---------------------------------------------------------------------------
__global__ __launch_bounds__(256) void gat_node_prep(
    const float* __restrict__ h,
    const float* __restrict__ att_src, const float* __restrict__ att_dst,
    float* __restrict__ a_src, float* __restrict__ a_dst,
    unsigned* __restrict__ mmax, float* __restrict__ denom,
    float* __restrict__ num, int n)
{
    const int wave = threadIdx.x >> 5;
    const int lane = threadIdx.x & 31;
    const int node = blockIdx.x * 8 + wave;
    if (node >= n) return;

    const float* __restrict__ hp = h + (size_t)node * HID;
    const float h0 = hp[lane], h1 = hp[32 + lane];
    float as = h0 * att_src[lane] + h1 * att_src[32 + lane];
    float ad = h0 * att_dst[lane] + h1 * att_dst[32 + lane];
    #pragma unroll
    for (int off = 16; off; off >>= 1) {
        as += __shfl_xor(as, off, 32);
        ad += __shfl_xor(ad, off, 32);
    }
    if (lane == 0) {
        a_src[node] = as;
        a_dst[node] = ad;
        mmax[node]  = 0u;      // below mapped(-inf)
        denom[node] = 0.0f;
    }
    float* __restrict__ np = num + (size_t)node * HID;
    np[lane] = 0.0f;
    np[32 + lane] = 0.0f;
}

// order-preserving float<->uint map for atomicMax on floats
__device__ __forceinline__ unsigned fmap(float f) {
    unsigned u = __float_as_uint(f);
    return (u & 0x80000000u) ? ~u : (u | 0x80000000u);
}
__device__ __forceinline__ float funmap(unsigned u) {
    return __uint_as_float((u & 0x80000000u) ? (u ^ 0x80000000u) : ~u);
}

// ---------------------------------------------------------------------------
// Kernel 3: segment max of leaky-relu logits (edges + self loops).
// ---------------------------------------------------------------------------
__global__ __launch_bounds__(256) void gat_edge_max(
    const int* __restrict__ src, const int* __restrict__ dst,
    const float* __restrict__ a_src, const float* __restrict__ a_dst,
    unsigned* __restrict__ mmax, int nedges, int n)
{
    const int e = blockIdx.x * blockDim.x + threadIdx.x;
    if (e >= nedges + n) return;
    int s, d;
    if (e < nedges) { s = src[e]; d = dst[e]; }
    else            { s = d = e - nedges; }
    float v = a_src[s] + a_dst[d];
    v = v > 0.0f ? v : NEG_SLOPE * v;
    atomicMax(mmax + d, fmap(v));
}

// ---------------------------------------------------------------------------
// Kernel 4: scatter pass — w = exp(logit - max[dst]); denom += w;
// num[dst][c] += w * h[src][c].  One wave per edge, coalesced 64-ch I/O.
// h (12.8MB) and num (12.8MB) are L2-resident on MI455X (192MB L2).
// ---------------------------------------------------------------------------
__global__ __launch_bounds__(256) void gat_edge_scatter(
    const int* __restrict__ src, const int* __restrict__ dst,
    const float* __restrict__ a_src, const float* __restrict__ a_dst,
    const unsigned* __restrict__ mmax, const float* __restrict__ h,
    float* __restrict__ denom, float* __restrict__ num, int nedges, int n)
{
    const int lane = threadIdx.x & 31;
    const int eIdx = (int)((blockIdx.x * (unsigned)blockDim.x + threadIdx.x) >> 5);
    if (eIdx >= nedges + n) return;
    int s, d;
    if (eIdx < nedges) { s = src[eIdx]; d = dst[eIdx]; }
    else               { s = d = eIdx - nedges; }
    float v = a_src[s] + a_dst[d];
    v = v > 0.0f ? v : NEG_SLOPE * v;
    const float w = __expf(v - funmap(mmax[d]));
    if (lane == 0) atomicAdd(denom + d, w);
    const float* __restrict__ hp = h + (size_t)s * HID;
    float* __restrict__ np = num + (size_t)d * HID;
    atomicAdd(np + lane,      w * hp[lane]);
    atomicAdd(np + 32 + lane, w * hp[32 + lane]);
}

// ---------------------------------------------------------------------------
// Kernel 5: finalize per node: out = elu(num/denom + bias); fold fc into
// s1[n] = out·fc_w[:64], s2[n] = out·fc_w[64:].  One wave per node.
// ---------------------------------------------------------------------------
__global__ __launch_bounds__(256) void gat_finalize(
    const float* __restrict__ num, const float* __restrict__ denom,
    const float* __restrict__ bias, const float* __restrict__ fc_w,
    float* __restrict__ s1, float* __restrict__ s2, int n)
{
    const int wave = threadIdx.x >> 5;
    const int lane = threadIdx.x & 31;
    const int node = blockIdx.x * 8 + wave;
    if (node >= n) return;

    const float inv = 1.0f / denom[node];   // self-loop guarantees denom > 0
    const float* __restrict__ np = num + (size_t)node * HID;
    float v0 = np[lane]      * inv + bias[lane];
    float v1 = np[32 + lane] * inv + bias[32 + lane];
    v0 = v0 > 0.0f ? v0 : (__expf(v0) - 1.0f);   // elu
    v1 = v1 > 0.0f ? v1 : (__expf(v1) - 1.0f);
    float p1 = v0 * fc_w[lane]      + v1 * fc_w[32 + lane];
    float p2 = v0 * fc_w[64 + lane] + v1 * fc_w[96 + lane];
    #pragma unroll
    for (int off = 16; off; off >>= 1) {
        p1 += __shfl_xor(p1, off, 32);
        p2 += __shfl_xor(p2, off, 32);
    }
    if (lane == 0) { s1[node] = p1; s2[node] = p2; }
}

// ---------------------------------------------------------------------------
// Kernel 6: edge scores = s1[src] + s2[dst] + fc_b  (original edges only).
// ---------------------------------------------------------------------------
__global__ __launch_bounds__(256) void gat_score(
    const int* __restrict__ src, const int* __restrict__ dst,
    const float* __restrict__ s1, const float* __restrict__ s2,
    const float* __restrict__ fc_b, float* __restrict__ out, int nedges)
{
    const int e = blockIdx.x * blockDim.x + threadIdx.x;
    if (e >= nedges) return;
    out[e] = s1[src[e]] + s2[dst[e]] + fc_b[0];
}

// ---------------------------------------------------------------------------
extern "C" void kernel_launch(void* const* d_in, const int* in_sizes, int n_in,
                              void* d_out, int out_size, void* d_ws, size_t ws_size,
                              hipStream_t stream)
{
    const float* x        = (const float*)d_in[0];
    const int*   eidx     = (const int*)  d_in[1];
    const float* W        = (const float*)d_in[2];
    const float* att_src  = (const float*)d_in[3];
    const float* att_dst  = (const float*)d_in[4];
    const float* bias     = (const float*)d_in[5];
    const float* fc_w     = (const float*)d_in[6];
    const float* fc_b     = (const float*)d_in[7];
    float* out            = (float*)d_out;

    const int N_ = in_sizes[0] / F_IN;      // 50000
    const int E_ = in_sizes[1] / 2;         // 800000
    const int* src = eidx;
    const int* dst = eidx + E_;

    // workspace carve-up (floats)
    float* ws   = (float*)d_ws;
    float*    h      = ws;                         // N*64
    float*    num    = h + (size_t)N_ * HID;       // N*64
    float*    a_src  = num + (size_t)N_ * HID;     // N
    float*    a_dst  = a_src + N_;                 // N
    float*    denom  = a_dst + N_;                 // N
    float*    s1     = denom + N_;                 // N
    float*    s2     = s1 + N_;                    // N
    unsigned* mmax   = (unsigned*)(s2 + N_);       // N

    const int total_e = E_ + N_;

    // 1) GEMM (WMMA): 128 rows / block
    gat_gemm_wmma<<<(N_ + 127) / 128, 256, 0, stream>>>(x, W, h, N_);
    // 2) node scalars + scratch init: 8 nodes / block
    gat_node_prep<<<(N_ + 7) / 8, 256, 0, stream>>>(h, att_src, att_dst,
                                                    a_src, a_dst, mmax, denom, num, N_);
    // 3) segment max
    gat_edge_max<<<(total_e + 255) / 256, 256, 0, stream>>>(src, dst, a_src, a_dst,
                                                            mmax, E_, N_);
    // 4) exp + scatter-add (1 wave per edge)
    {
        long long threads = (long long)total_e * 32;
        int blocks = (int)((threads + 255) / 256);
        gat_edge_scatter<<<blocks, 256, 0, stream>>>(src, dst, a_src, a_dst,
                                                     mmax, h, denom, num, E_, N_);
    }
    // 5) normalize + elu + fold fc weights per node
    gat_finalize<<<(N_ + 7) / 8, 256, 0, stream>>>(num, denom, bias, fc_w, s1, s2, N_);
    // 6) edge scores
    gat_score<<<(E_ + 255) / 256, 256, 0, stream>>>(src, dst, s1, s2, fc_b, out, E_);
}